// NeuralSpline_22419729285565
// MI455X (gfx1250) — compile-verified
//
#include <hip/hip_runtime.h>

typedef __attribute__((ext_vector_type(2))) float v2f;
typedef __attribute__((ext_vector_type(4))) float v4f;
typedef __attribute__((ext_vector_type(8))) float v8f;

// ---------------------------------------------------------------------------
// Zero-fill helper (stats + FC1 accumulators)
// ---------------------------------------------------------------------------
__global__ void zero_f32(float* p, int n) {
  int i = blockIdx.x * blockDim.x + threadIdx.x;
  if (i < n) p[i] = 0.f;
}

// ---------------------------------------------------------------------------
// Shared conv epilogue: bias + ReLU, NCHW store, BN sum/sumsq reduction.
// C/D layout: vgpr v -> channel ct*16 + (lane>>4)*8 + v, col = lane&15.
// ---------------------------------------------------------------------------
__device__ __forceinline__ void conv_epilogue(
    v8f acc, const float* __restrict__ Bias, float* __restrict__ Y,
    float* __restrict__ s1, float* __restrict__ s2,
    int ct, int half, int l16, bool nok, int bi, int rr, int Cout, int HW)
{
  for (int v = 0; v < 8; ++v) {
    int cm = ct * 16 + half * 8 + v;
    float val = fmaxf(acc[v] + Bias[cm], 0.f);
    float sv = nok ? val : 0.f;
    if (nok) Y[((size_t)bi * Cout + cm) * HW + rr] = val;
    float s = sv, q = sv * sv;
    s += __shfl_xor(s, 1);  q += __shfl_xor(q, 1);
    s += __shfl_xor(s, 2);  q += __shfl_xor(q, 2);
    s += __shfl_xor(s, 4);  q += __shfl_xor(q, 4);
    s += __shfl_xor(s, 8);  q += __shfl_xor(q, 8);
    if (l16 == 0) { atomicAdd(s1 + cm, s); atomicAdd(s2 + cm, q); }
  }
}

// ---------------------------------------------------------------------------
// Layer-1 conv (Cin=3, K=27): incremental-tracker implicit GEMM (K not a
// multiple of 4, so the constant-stride tap-major layout doesn't apply).
// ---------------------------------------------------------------------------
__global__ void conv_bn_stats(const float* __restrict__ X,
                              const float* __restrict__ Wt,
                              const float* __restrict__ Bias,
                              float* __restrict__ Y,
                              float* __restrict__ s1,
                              float* __restrict__ s2,
                              int Cin, int Hin, int Win,
                              int Cout, int Hout, int Wout,
                              int tilesN, int tilesC)
{
  const int B = 32;
  int wave = (blockIdx.x * blockDim.x + threadIdx.x) >> 5;
  int lane = threadIdx.x & 31;
  int ct = wave / tilesN;
  int pt = wave - ct * tilesN;
  if (ct >= tilesC) return;          // wave-uniform

  const int K      = Cin * 9;
  const int K4     = K & ~3;
  const int HW     = Hout * Wout;
  const int Ntot   = B * HW;
  const int HinWin = Hin * Win;
  int half = lane >> 4;
  int l16  = lane & 15;

  int co = ct * 16 + l16;
  int n  = pt * 16 + l16;
  bool nok = (n < Ntot);
  int nc = nok ? n : (Ntot - 1);     // clamp: OOB lanes read valid memory
  int bi = nc / HW;
  int rr = nc - bi * HW;
  int oh = rr / Wout;
  int ow = rr - oh * Wout;
  const float* Xb = X + ((size_t)bi * Cin) * HinWin + (size_t)(oh * 2) * Win + (ow * 2);
  const float* Wp = Wt + (size_t)co * K + half * 2;

  int rem0 = half * 2;
  const float* Xc0 = Xb;

  v8f acc = {};
  #pragma unroll 2
  for (int k0 = 0; k0 < K4; k0 += 4) {
    v2f a, b;
    a.x = Wp[0];
    a.y = Wp[1];
    Wp += 4;
    int r0 = (rem0 * 11) >> 5;
    b.x = Xc0[r0 * Win + (rem0 - 3 * r0)];
    int rem1 = rem0 + 1;
    bool w1 = (rem1 == 9);
    rem1 = w1 ? 0 : rem1;
    const float* Xc1 = Xc0 + (w1 ? HinWin : 0);
    int r1 = (rem1 * 11) >> 5;
    b.y = Xc1[r1 * Win + (rem1 - 3 * r1)];
    rem0 += 4;
    bool w0 = (rem0 >= 9);
    rem0 = w0 ? rem0 - 9 : rem0;
    Xc0 += w0 ? HinWin : 0;
    acc = __builtin_amdgcn_wmma_f32_16x16x4_f32(false, a, false, b,
                                                (short)0, acc, false, false);
  }
  if (K4 < K) {                      // guarded tail (K=27)
    int ka = K4 + half * 2;
    v2f a, b;
    a.x = 0.f; a.y = 0.f; b.x = 0.f; b.y = 0.f;
    if (ka < K) {
      a.x = Wt[(size_t)co * K + ka];
      int ci = ka / 9, rem = ka - ci * 9;
      int r = rem / 3, sx = rem - r * 3;
      b.x = Xb[((size_t)ci * HinWin) + r * Win + sx];
    }
    if (ka + 1 < K) {
      int k1 = ka + 1;
      a.y = Wt[(size_t)co * K + k1];
      int ci = k1 / 9, rem = k1 - ci * 9;
      int r = rem / 3, sx = rem - r * 3;
      b.y = Xb[((size_t)ci * HinWin) + r * Win + sx];
    }
    acc = __builtin_amdgcn_wmma_f32_16x16x4_f32(false, a, false, b,
                                                (short)0, acc, false, false);
  }
  conv_epilogue(acc, Bias, Y, s1, s2, ct, half, l16, nok, bi, rr, Cout, HW);
}

// ---------------------------------------------------------------------------
// Layers 2-5: tap-major K ordering (k = tap*Cin + ci). Legal because GEMM K
// order is a free permutation when A and B use the same lane->K map. Both
// operands then advance with COMPILE-TIME constant strides:
//   A: W[co][ci][tap]        -> ptr += 36 floats / K-step, pair at +9
//   B: X[bi][ci][ih][iw]     -> ptr += 4*HIN*WIN / K-step, pair at +HIN*WIN
// No per-iteration index math beyond two pointer adds.
// ---------------------------------------------------------------------------
template<int CIN, int HIN, int WIN, int COUT, int HOUT, int WOUT>
__global__ void conv_tap(const float* __restrict__ X,
                         const float* __restrict__ Wt,
                         const float* __restrict__ Bias,
                         float* __restrict__ Y,
                         float* __restrict__ s1,
                         float* __restrict__ s2,
                         int tilesN)
{
  constexpr int K      = CIN * 9;
  constexpr int HW     = HOUT * WOUT;
  constexpr int Ntot   = 32 * HW;
  constexpr int HinWin = HIN * WIN;
  constexpr int tilesC = COUT / 16;

  int wave = (blockIdx.x * blockDim.x + threadIdx.x) >> 5;
  int lane = threadIdx.x & 31;
  int ct = wave / tilesN;
  int pt = wave - ct * tilesN;
  if (ct >= tilesC) return;          // wave-uniform

  int half = lane >> 4;
  int l16  = lane & 15;
  int co = ct * 16 + l16;
  int n  = pt * 16 + l16;
  bool nok = (n < Ntot);
  int nc = nok ? n : (Ntot - 1);
  int bi = nc / HW;
  int rr = nc - bi * HW;
  int oh = rr / WOUT;
  int ow = rr - oh * WOUT;
  const float* Xb = X + ((size_t)bi * CIN) * HinWin + (size_t)(oh * 2) * WIN + (ow * 2)
                      + (size_t)(half * 2) * HinWin;
  const float* Wb = Wt + (size_t)co * K + half * 2 * 9;

  v8f acc = {};
  #pragma unroll
  for (int tap = 0; tap < 9; ++tap) {
    constexpr int dummy = 0; (void)dummy;
    const int r = tap / 3, sx = tap % 3;          // compile-time (unrolled)
    const float* Xp = Xb + r * WIN + sx;
    const float* Wp = Wb + tap;
    #pragma unroll 2
    for (int c0 = 0; c0 < CIN; c0 += 4) {
      v2f a, b;
      a.x = Wp[0];
      a.y = Wp[9];
      b.x = Xp[0];
      b.y = Xp[HinWin];
      Wp += 36;
      Xp += (size_t)4 * HinWin;
      acc = __builtin_amdgcn_wmma_f32_16x16x4_f32(false, a, false, b,
                                                  (short)0, acc, false, false);
    }
  }
  conv_epilogue(acc, Bias, Y, s1, s2, ct, half, l16, nok, bi, rr, COUT, HW);
}

// ---------------------------------------------------------------------------
// Batch-norm apply (in place)
// ---------------------------------------------------------------------------
__global__ void bn_apply(float* __restrict__ Y,
                         const float* __restrict__ s1, const float* __restrict__ s2,
                         const float* __restrict__ g, const float* __restrict__ be,
                         int C, int HW, int total, float invCnt)
{
  int i = blockIdx.x * blockDim.x + threadIdx.x;
  if (i >= total) return;
  int c = (i / HW) % C;
  float m   = s1[c] * invCnt;
  float var = s2[c] * invCnt - m * m;
  float inv = rsqrtf(var + 1e-5f);
  Y[i] = g[c] * (Y[i] - m) * inv + be[c];
}

// ---------------------------------------------------------------------------
// FC1 partial GEMM: M=1000, N=32, K=25088 split 8 ways (blockIdx.y).
// K stepped by 8: one b128 load per operand feeds two WMMAs. Prefetch runs
// ahead on the 100MB weight stream (global_prefetch_b8).
// ---------------------------------------------------------------------------
__global__ void fc1_wmma(const float* __restrict__ Xa,   // [32][25088]
                         const float* __restrict__ Wt,   // [1000][25088]
                         float* __restrict__ O,          // [1000][32] accum
                         int kchunk)
{
  const int K = 25088, M = 1000;
  int wave = (blockIdx.x * blockDim.x + threadIdx.x) >> 5;
  int lane = threadIdx.x & 31;
  int mt = wave >> 1, nt = wave & 1;
  if (mt >= 63) return;                      // wave-uniform
  int half = lane >> 4, l16 = lane & 15;
  int m = mt * 16 + l16;
  int n = nt * 16 + l16;                     // always < 32
  int mc = m < M ? m : (M - 1);              // clamp (rows >=1000 masked later)
  const float* Wr = Wt + (size_t)mc * K + half * 4;
  const float* Xr = Xa + (size_t)n * K + half * 4;
  int kbeg = blockIdx.y * kchunk;
  int kend = kbeg + kchunk;
  v8f acc = {};
  for (int k0 = kbeg; k0 < kend; k0 += 8) {
    __builtin_prefetch(Wr + k0 + 512, 0, 1); // run ahead on weight stream
    v4f a4 = *(const v4f*)(Wr + k0);         // 16B aligned
    v4f b4 = *(const v4f*)(Xr + k0);
    v2f a, b;
    a.x = a4.x; a.y = a4.y; b.x = b4.x; b.y = b4.y;
    acc = __builtin_amdgcn_wmma_f32_16x16x4_f32(false, a, false, b,
                                                (short)0, acc, false, false);
    a.x = a4.z; a.y = a4.w; b.x = b4.z; b.y = b4.w;
    acc = __builtin_amdgcn_wmma_f32_16x16x4_f32(false, a, false, b,
                                                (short)0, acc, false, false);
  }
  for (int v = 0; v < 8; ++v) {
    int mm = mt * 16 + half * 8 + v;
    if (mm < M) atomicAdd(O + mm * 32 + nt * 16 + l16, acc[v]);
  }
}

__global__ void fc1_finish(float* __restrict__ O, const float* __restrict__ Bias) {
  int i = blockIdx.x * blockDim.x + threadIdx.x;
  if (i < 32000) O[i] = fmaxf(O[i] + Bias[i >> 5], 0.f);
}

// ---------------------------------------------------------------------------
// FC2 + spline matrix product + cubic coefficients (96 tiny problems).
// ---------------------------------------------------------------------------
__global__ void head_spline_coef(const float* __restrict__ O1,  // [1000][32]
                                 const float* __restrict__ W2,  // [30][1000]
                                 const float* __restrict__ B2,  // [30]
                                 const float* __restrict__ Mat, // [10][10]
                                 float* __restrict__ coef)      // [96][36]
{
  int t = threadIdx.x;
  if (t >= 96) return;
  int b = t / 3, c = t - 3 * (t / 3);
  float ys[10];
  for (int i = 0; i < 10; ++i) {
    int j = c * 10 + i;
    float acc = B2[j];
    const float* wr = W2 + (size_t)j * 1000;
    for (int p = 0; p < 1000; ++p) acc += wr[p] * O1[p * 32 + b];
    ys[i] = acc * 0.01f + (float)i * (1.f / 9.f);    // /100 + identity
  }
  float Mm[10];
  for (int i = 0; i < 10; ++i) {
    float a = 0.f;
    for (int j = 0; j < 10; ++j) a += Mat[i * 10 + j] * ys[j];
    Mm[i] = a;
  }
  const float h = 1.f / 9.f;
  float* cf = coef + t * 36;
  for (int i = 0; i < 9; ++i) {
    cf[i]      = (Mm[i + 1] - Mm[i]) / (6.f * h);
    cf[9 + i]  = Mm[i] * 0.5f;
    cf[18 + i] = (ys[i + 1] - ys[i]) / h - (Mm[i + 1] + 2.f * Mm[i]) * (h / 6.f);
    cf[27 + i] = ys[i];
  }
}

// ---------------------------------------------------------------------------
// Streaming spline evaluation (float4 / b128): 25MB in + 25MB out.
// ---------------------------------------------------------------------------
__device__ __forceinline__ float eval1(const float* cf, float xv) {
  int xi = (int)floorf(xv * 9.f);
  xi = xi < 0 ? 0 : (xi > 8 ? 8 : xi);
  float xf = xv - (float)xi * (1.f / 9.f);
  return ((cf[xi] * xf + cf[9 + xi]) * xf + cf[18 + xi]) * xf + cf[27 + xi];
}

__global__ void spline_eval(const float* __restrict__ batch,
                            const float* __restrict__ coef,
                            float* __restrict__ out,
                            float* __restrict__ spl)
{
  __shared__ float cf[36];
  int bc = blockIdx.x;
  if (threadIdx.x < 36) cf[threadIdx.x] = coef[bc * 36 + threadIdx.x];
  __syncthreads();
  const float* in = batch + (size_t)bc * 65536;
  float* o = out + (size_t)bc * 65536;
  int chunk = 65536 / gridDim.y;
  int start = blockIdx.y * chunk;
  int end   = start + chunk;
  for (int i = start + threadIdx.x * 4; i < end; i += blockDim.x * 4) {
    v4f x4 = *(const v4f*)(in + i);
    v4f r;
    r.x = eval1(cf, x4.x);
    r.y = eval1(cf, x4.y);
    r.z = eval1(cf, x4.z);
    r.w = eval1(cf, x4.w);
    *(v4f*)(o + i) = r;
  }
  if (blockIdx.y == 0) {
    for (int i = threadIdx.x; i < 255; i += blockDim.x)
      spl[bc * 255 + i] = eval1(cf, (float)i * (1.f / 255.f));
  }
}

// ---------------------------------------------------------------------------
// Host-side orchestration (graph-capture safe)
// ---------------------------------------------------------------------------
extern "C" void kernel_launch(void* const* d_in, const int* in_sizes, int n_in,
                              void* d_out, int out_size, void* d_ws, size_t ws_size,
                              hipStream_t stream)
{
  const float* batch = (const float*)d_in[0];
  const float* cw[5] = {(const float*)d_in[1],  (const float*)d_in[5],
                        (const float*)d_in[9],  (const float*)d_in[13],
                        (const float*)d_in[17]};
  const float* cb[5] = {(const float*)d_in[2],  (const float*)d_in[6],
                        (const float*)d_in[10], (const float*)d_in[14],
                        (const float*)d_in[18]};
  const float* bg[5] = {(const float*)d_in[3],  (const float*)d_in[7],
                        (const float*)d_in[11], (const float*)d_in[15],
                        (const float*)d_in[19]};
  const float* bb[5] = {(const float*)d_in[4],  (const float*)d_in[8],
                        (const float*)d_in[12], (const float*)d_in[16],
                        (const float*)d_in[20]};
  const float* l1w = (const float*)d_in[21];
  const float* l1b = (const float*)d_in[22];
  const float* l2w = (const float*)d_in[23];
  const float* l2b = (const float*)d_in[24];
  const float* mat = (const float*)d_in[25];

  // Workspace layout (floats): stats | coef | O1 | ping (16.5M) | pong (8.2M)
  float* ws    = (float*)d_ws;
  float* stats = ws;
  float* coef  = ws + 2048;
  float* O1    = ws + 8192;
  float* bufA  = ws + 40960;
  float* bufB  = bufA + 16516096;

  zero_f32<<<160, 256, 0, stream>>>(ws, 40960);   // stats + coef + O1

  const int Hout[5] = {127, 63, 31, 15, 7};
  const int Cout[5] = {32, 64, 128, 256, 512};
  const int soff[5] = {0, 32, 96, 224, 480};
  float* ybuf[5] = {bufA, bufB, bufA, bufB, bufA};

  // Per-layer wave counts: tilesN = ceil(32*Ho*Wo/16), tilesC = Cout/16.
  int tilesN[5], blocks[5];
  for (int i = 0; i < 5; ++i) {
    int Ntot = 32 * Hout[i] * Hout[i];
    tilesN[i] = (Ntot + 15) / 16;
    blocks[i] = (tilesN[i] * (Cout[i] / 16) + 7) / 8;
  }

  // Layer 1 (Cin=3): tracker kernel.
  conv_bn_stats<<<blocks[0], 256, 0, stream>>>(
      batch, cw[0], cb[0], ybuf[0], stats + soff[0], stats + 992 + soff[0],
      3, 256, 256, 32, 127, 127, tilesN[0], 2);
  bn_apply<<<(32 * 32 * 127 * 127 + 255) / 256, 256, 0, stream>>>(
      ybuf[0], stats + soff[0], stats + 992 + soff[0], bg[0], bb[0],
      32, 127 * 127, 32 * 32 * 127 * 127, 1.f / (32.f * 127 * 127));

  // Layers 2-5: templated tap-major kernels (constant-stride inner loops).
  conv_tap<32, 127, 127, 64, 63, 63><<<blocks[1], 256, 0, stream>>>(
      ybuf[0], cw[1], cb[1], ybuf[1], stats + soff[1], stats + 992 + soff[1], tilesN[1]);
  bn_apply<<<(32 * 64 * 63 * 63 + 255) / 256, 256, 0, stream>>>(
      ybuf[1], stats + soff[1], stats + 992 + soff[1], bg[1], bb[1],
      64, 63 * 63, 32 * 64 * 63 * 63, 1.f / (32.f * 63 * 63));

  conv_tap<64, 63, 63, 128, 31, 31><<<blocks[2], 256, 0, stream>>>(
      ybuf[1], cw[2], cb[2], ybuf[2], stats + soff[2], stats + 992 + soff[2], tilesN[2]);
  bn_apply<<<(32 * 128 * 31 * 31 + 255) / 256, 256, 0, stream>>>(
      ybuf[2], stats + soff[2], stats + 992 + soff[2], bg[2], bb[2],
      128, 31 * 31, 32 * 128 * 31 * 31, 1.f / (32.f * 31 * 31));

  conv_tap<128, 31, 31, 256, 15, 15><<<blocks[3], 256, 0, stream>>>(
      ybuf[2], cw[3], cb[3], ybuf[3], stats + soff[3], stats + 992 + soff[3], tilesN[3]);
  bn_apply<<<(32 * 256 * 15 * 15 + 255) / 256, 256, 0, stream>>>(
      ybuf[3], stats + soff[3], stats + 992 + soff[3], bg[3], bb[3],
      256, 15 * 15, 32 * 256 * 15 * 15, 1.f / (32.f * 15 * 15));

  conv_tap<256, 15, 15, 512, 7, 7><<<blocks[4], 256, 0, stream>>>(
      ybuf[3], cw[4], cb[4], ybuf[4], stats + soff[4], stats + 992 + soff[4], tilesN[4]);
  bn_apply<<<(32 * 512 * 7 * 7 + 255) / 256, 256, 0, stream>>>(
      ybuf[4], stats + soff[4], stats + 992 + soff[4], bg[4], bb[4],
      512, 7 * 7, 32 * 512 * 7 * 7, 1.f / (32.f * 7 * 7));

  // FC1: 8-way K split (25088/8 = 3136), 1008 active waves.
  fc1_wmma<<<dim3(16, 8), 256, 0, stream>>>(ybuf[4], l1w, O1, 25088 / 8);
  fc1_finish<<<125, 256, 0, stream>>>(O1, l1b);
  head_spline_coef<<<1, 128, 0, stream>>>(O1, l2w, l2b, mat, coef);

  float* outp = (float*)d_out;
  spline_eval<<<dim3(96, 16), 256, 0, stream>>>(batch, coef, outp, outp + 6291456);
}